// KVCacheModule_66529043415042
// MI455X (gfx1250) — compile-verified
//
#include <hip/hip_runtime.h>

// KV-cache update for MI455X (gfx1250).
// out[b,s,h,:] = past[b,h,s,:]; then out[b, cache_id[b]+i, h, :] = key[b,h,i,:].
// Pure data movement: 1 GiB read + 1 GiB write => ~92us floor at 23.3 TB/s.
// D=128 floats (512 B) rows stay contiguous under the transpose; each wave32
// moves rows with per-lane 128-bit accesses, fully coalesced both ways.
// Bulk copy is staged through LDS with the CDNA5 async data-mover
// (GLOBAL_LOAD_ASYNC_TO_LDS_B128 / GLOBAL_STORE_ASYNC_FROM_LDS_B128, ASYNCcnt).
// Within a wave, 16 consecutive rows live in one (b,h) panel, so src stride is
// a constant 512 B and dst stride a constant 8192 B: one base address each,
// constant-offset GEPs per row (foldable into IOFFSET / single v_add), no
// per-row guards or 64-bit math in the hot path.

typedef __attribute__((ext_vector_type(4))) float v4f;

// The async-LDS b128 builtins take GCC-vector int4 pointers:
//   (v4i addrspace(1)* global, v4i addrspace(3)* lds, Imm offset, Imm cpol)
typedef int v4i __attribute__((vector_size(16)));
typedef __attribute__((address_space(1))) v4i* g_v4i_ptr;
typedef __attribute__((address_space(3))) v4i* l_v4i_ptr;

namespace {
constexpr unsigned kH = 16;             // heads
constexpr unsigned kSmax = 8192;        // max sequence length
constexpr unsigned kD = 128;            // head dim in floats (512-byte rows)
constexpr unsigned kWavesPerBlock = 8;  // 256 threads / wave32
constexpr unsigned kRowsPerWave = 16;   // 8 KB staged per wave (64 KB LDS/block)
}  // namespace

#if defined(__has_builtin)
#if __has_builtin(__builtin_amdgcn_global_load_async_to_lds_b128) && \
    __has_builtin(__builtin_amdgcn_global_store_async_from_lds_b128)
#define KV_USE_ASYNC_LDS 1
#endif
#if __has_builtin(__builtin_amdgcn_s_wait_asynccnt)
#define KV_HAVE_WAIT_ASYNC_BUILTIN 1
#endif
#endif

__device__ __forceinline__ void kv_wait_async0() {
#ifdef KV_HAVE_WAIT_ASYNC_BUILTIN
  __builtin_amdgcn_s_wait_asynccnt(0);
#else
  asm volatile("s_wait_asynccnt 0x0" ::: "memory");
#endif
}

// past row index: row = ((b*16 + h) << 13) + s ; out row index: ((b<<13)|s)<<4 | h
__device__ __forceinline__ unsigned kv_dst_off(unsigned row, unsigned lane) {
  const unsigned b = row >> 17;
  const unsigned h = (row >> 13) & (kH - 1);
  const unsigned s = row & (kSmax - 1);
  const unsigned drow = (((b << 13) | s) << 4) | h;
  return drow * kD + lane * 4u;  // float offset, max 2^28-1, fits in u32
}

__global__ __launch_bounds__(256) void kv_transpose_copy(const float* __restrict__ past,
                                                         float* __restrict__ out,
                                                         unsigned rows_total) {
  const unsigned wave_in_blk = threadIdx.x >> 5;
  const unsigned lane = threadIdx.x & 31;
  const unsigned gwave = blockIdx.x * kWavesPerBlock + wave_in_blk;
  const unsigned row0 = gwave * kRowsPerWave;

#ifdef KV_USE_ASYNC_LDS
  __shared__ char smem[kWavesPerBlock * kRowsPerWave * 512];
  char* slice = smem + wave_in_blk * (kRowsPerWave * 512) + lane * 16u;

  if (row0 + kRowsPerWave <= rows_total) {
    // Fast path (always taken at the reference shapes): one base per side,
    // constant strides: src +512 B/row, dst +8192 B/row, LDS +512 B/row.
    const unsigned b = row0 >> 17;
    const unsigned h = (row0 >> 13) & (kH - 1);
    const unsigned s0 = row0 & (kSmax - 1);  // s0..s0+15 stay inside the panel
    const char* gsrc =
        (const char*)past + (unsigned long long)row0 * 512ull + lane * 16u;
    char* gdst = (char*)out +
                 (unsigned long long)((((b << 13) | s0) << 4) | h) * 512ull +
                 lane * 16u;
#pragma unroll
    for (unsigned r = 0; r < kRowsPerWave; ++r) {
      __builtin_amdgcn_global_load_async_to_lds_b128(
          (g_v4i_ptr)(void*)(gsrc + r * 512u), (l_v4i_ptr)(void*)(slice + r * 512u), 0, 0);
    }
    kv_wait_async0();  // LDS writes from async loads must land before async stores read
#pragma unroll
    for (unsigned r = 0; r < kRowsPerWave; ++r) {
      __builtin_amdgcn_global_store_async_from_lds_b128(
          (g_v4i_ptr)(void*)(gdst + r * 8192u), (l_v4i_ptr)(void*)(slice + r * 512u), 0, 0);
    }
  } else {
    // Tail path (unreachable at the reference shapes); per-row guards.
#pragma unroll
    for (unsigned r = 0; r < kRowsPerWave; ++r) {
      const unsigned row = row0 + r;
      if (row < rows_total) {
        const float* src = past + (unsigned long long)row * kD + lane * 4u;
        __builtin_amdgcn_global_load_async_to_lds_b128(
            (g_v4i_ptr)(void*)src, (l_v4i_ptr)(void*)(slice + r * 512u), 0, 0);
      }
    }
    kv_wait_async0();
#pragma unroll
    for (unsigned r = 0; r < kRowsPerWave; ++r) {
      const unsigned row = row0 + r;
      if (row < rows_total) {
        float* dst = out + kv_dst_off(row, lane);
        __builtin_amdgcn_global_store_async_from_lds_b128(
            (g_v4i_ptr)(void*)dst, (l_v4i_ptr)(void*)(slice + r * 512u), 0, 0);
      }
    }
  }
  // S_ENDPGM performs an implicit wait-idle, draining ASYNCcnt before wave exit.
#else
  // Fallback: direct VGPR streaming copy, nontemporal b128 both ways.
  v4f vals[kRowsPerWave];
#pragma unroll
  for (unsigned r = 0; r < kRowsPerWave; ++r) {
    const unsigned row = row0 + r;
    if (row < rows_total) {
      const v4f* src = (const v4f*)(past + (unsigned long long)row * kD + lane * 4u);
      vals[r] = __builtin_nontemporal_load(src);
    }
  }
#pragma unroll
  for (unsigned r = 0; r < kRowsPerWave; ++r) {
    const unsigned row = row0 + r;
    if (row < rows_total) {
      v4f* dst = (v4f*)(out + kv_dst_off(row, lane));
      __builtin_nontemporal_store(vals[r], dst);
    }
  }
#endif
}

// key rows = (b*16 + h)*16 + i; one wave per 512-byte row (4096 rows, 2 MiB).
__global__ __launch_bounds__(256) void kv_scatter_keys(const float* __restrict__ key,
                                                       const int* __restrict__ cache_id,
                                                       const int* __restrict__ batch_id,
                                                       float* __restrict__ out,
                                                       unsigned rows_total) {
  const unsigned gwave = (blockIdx.x * blockDim.x + threadIdx.x) >> 5;
  const unsigned lane = threadIdx.x & 31;
  if (gwave >= rows_total) return;
  const unsigned b = gwave >> 8;
  const unsigned h = (gwave >> 4) & 15;
  const unsigned i = gwave & 15;
  const unsigned s = (unsigned)cache_id[b] + i;          // < S_max by construction
  const unsigned ob = (unsigned)batch_id[b];             // identity, kept for fidelity
  const v4f v = *(const v4f*)(key + (unsigned long long)gwave * kD + lane * 4u);
  const unsigned drow = (((ob << 13) | (s & (kSmax - 1))) << 4) | h;
  *(v4f*)(out + (unsigned long long)drow * kD + lane * 4u) = v;
}

extern "C" void kernel_launch(void* const* d_in, const int* in_sizes, int n_in,
                              void* d_out, int out_size, void* d_ws, size_t ws_size,
                              hipStream_t stream) {
  const int*   cache_id = (const int*)d_in[0];
  const int*   batch_id = (const int*)d_in[1];
  const float* key      = (const float*)d_in[2];
  const float* past     = (const float*)d_in[3];
  float*       out      = (float*)d_out;

  // 1) Bulk transpose-copy of `past` into `out`: 2,097,152 rows of 512 B.
  const unsigned copy_rows = (unsigned)((unsigned long long)in_sizes[3] / kD);
  const unsigned rows_per_block = kWavesPerBlock * kRowsPerWave;  // 128
  const unsigned copy_blocks = (copy_rows + rows_per_block - 1) / rows_per_block;
  kv_transpose_copy<<<dim3(copy_blocks), dim3(256), 0, stream>>>(past, out, copy_rows);

  // 2) Overwrite the 16 new rows per (b, h) with `key` (stream-ordered after the copy).
  const unsigned key_rows = (unsigned)((unsigned long long)in_sizes[2] / kD);  // 4096
  const unsigned key_blocks = (key_rows + kWavesPerBlock - 1) / kWavesPerBlock;
  kv_scatter_keys<<<dim3(key_blocks), dim3(256), 0, stream>>>(key, cache_id, batch_id, out,
                                                              key_rows);
}